// AttentiveFPModel_V1_16020228014644
// MI455X (gfx1250) — compile-verified
//
#include <hip/hip_runtime.h>
#include <hip/hip_bf16.h>
#include <math.h>

// ---------------- problem constants (match reference) ----------------
#define NN   50000
#define EE_  800000
#define CC   64
#define HH   192
#define EDD  16
#define GG   2048

typedef __attribute__((ext_vector_type(2))) float v2f;
typedef __attribute__((ext_vector_type(8))) float v8f;

// ---------------- helpers ----------------
__device__ __forceinline__ unsigned fkey(float f) {
    unsigned b = __float_as_uint(f);
    return (b & 0x80000000u) ? ~b : (b | 0x80000000u);
}
__device__ __forceinline__ float fdec(unsigned k) {
    unsigned b = (k & 0x80000000u) ? (k & 0x7fffffffu) : ~k;
    return __uint_as_float(b);
}
__device__ __forceinline__ float wave_sum(float v) {
    for (int off = 16; off; off >>= 1) v += __shfl_down(v, off, 32);
    return v;
}

// ---------------- WMMA fp32 GEMM:  Y[M,Nout] = X[M,K] @ W[Nout,K]^T (+bias)(+act) ----
// One 16x64 output strip per wave: 4 accumulators over 4 adjacent n-tiles share one
// A fragment per K-step -> 5 b64 loads per 4 V_WMMA_F32_16X16X4_F32 (vs 8 unblocked).
// A (16x4, 32-bit): lane<16 holds M=lane, {K=k0,k0+1}; lane>=16 holds M=lane-16,{K=k0+2,k0+3}.
// B (4x16) mirrors A on W rows (Y = X W^T symmetry).
// Requires Nout % 64 == 0 (true: 192, 576). act: 0 = none, 1 = leaky_relu(0.01).
__global__ void k_gemm_wmma(const float* __restrict__ X, const float* __restrict__ W,
                            const float* __restrict__ bias, float* __restrict__ Y,
                            int M, int K, int Nout, int act)
{
    int lane = threadIdx.x & 31;
    int wv   = threadIdx.x >> 5;
    int mt   = blockIdx.x * 4 + wv;          // m-tile per wave
    if (mt * 16 >= M) return;                // wave-uniform exit; EXEC stays full below
    int m0 = mt * 16;
    int n0 = blockIdx.y * 64;                // 4 n-tiles per wave
    int hf = lane >> 4;                      // half: 0 or 1
    int l  = lane & 15;

    const float* xr = X + (size_t)(m0 + l) * K + 2 * hf;
    const float* wr = W + (size_t)(n0 + l) * K + 2 * hf;
    const size_t ws = (size_t)16 * K;        // stride between n-tiles in W

    v8f acc0 = {0.f,0.f,0.f,0.f,0.f,0.f,0.f,0.f};
    v8f acc1 = acc0, acc2 = acc0, acc3 = acc0;

    #pragma unroll 2
    for (int k0 = 0; k0 < K; k0 += 4) {
        v2f a, b0, b1, b2, b3;
        a.x  = xr[k0];            a.y  = xr[k0 + 1];
        b0.x = wr[k0];            b0.y = wr[k0 + 1];
        b1.x = wr[ws + k0];       b1.y = wr[ws + k0 + 1];
        b2.x = wr[2 * ws + k0];   b2.y = wr[2 * ws + k0 + 1];
        b3.x = wr[3 * ws + k0];   b3.y = wr[3 * ws + k0 + 1];
        acc0 = __builtin_amdgcn_wmma_f32_16x16x4_f32(false, a, false, b0, (short)0, acc0, false, false);
        acc1 = __builtin_amdgcn_wmma_f32_16x16x4_f32(false, a, false, b1, (short)0, acc1, false, false);
        acc2 = __builtin_amdgcn_wmma_f32_16x16x4_f32(false, a, false, b2, (short)0, acc2, false, false);
        acc3 = __builtin_amdgcn_wmma_f32_16x16x4_f32(false, a, false, b3, (short)0, acc3, false, false);
    }

    int row0 = m0 + hf * 8;
    union { v8f v; float f[8]; } u0, u1, u2, u3;
    u0.v = acc0; u1.v = acc1; u2.v = acc2; u3.v = acc3;
    float bc0 = bias ? bias[n0 + l]      : 0.f;
    float bc1 = bias ? bias[n0 + 16 + l] : 0.f;
    float bc2 = bias ? bias[n0 + 32 + l] : 0.f;
    float bc3 = bias ? bias[n0 + 48 + l] : 0.f;
    #pragma unroll
    for (int r = 0; r < 8; ++r) {
        float* yr = Y + (size_t)(row0 + r) * Nout + n0 + l;
        float v0 = u0.f[r] + bc0, v1 = u1.f[r] + bc1;
        float v2 = u2.f[r] + bc2, v3 = u3.f[r] + bc3;
        if (act == 1) {
            v0 = (v0 > 0.f) ? v0 : 0.01f * v0;
            v1 = (v1 > 0.f) ? v1 : 0.01f * v1;
            v2 = (v2 > 0.f) ? v2 : 0.01f * v2;
            v3 = (v3 > 0.f) ? v3 : 0.01f * v3;
        }
        yr[0] = v0; yr[16] = v1; yr[32] = v2; yr[48] = v3;
    }
}

// ---------------- small kernels ----------------

// ve[j] = sum_k att_e[k] * We[k*ED + j]      (We is [H, ED] row-major)
__global__ void k_wevec(const float* __restrict__ We, const float* __restrict__ att_e,
                        float* __restrict__ ve)
{
    int j = threadIdx.x;
    if (j >= EDD) return;
    float s = 0.f;
    for (int k = 0; k < HH; ++k) s += att_e[k] * We[k * EDD + j];
    ve[j] = s;
}

// per-edge alpha_e = edge_attr[e] . ve ; also accumulate sum for the self-loop mean term
__global__ void k_edgedot(const float* __restrict__ ea, const float* __restrict__ ve,
                          float* __restrict__ aE, float* __restrict__ sumb)
{
    int e = blockIdx.x * blockDim.x + threadIdx.x;
    float dot = 0.f;
    if (e < EE_) {
        const float* r = ea + (size_t)e * EDD;
        #pragma unroll
        for (int j = 0; j < EDD; ++j) dot += r[j] * ve[j];
        aE[e] = dot;
    }
    float w = wave_sum(dot);
    if ((threadIdx.x & 31) == 0) atomicAdd(sumb, w);
}

// per-row dots: s[i] = row.a1 ; (optional) d[i] = row.a2 ; one wave per row
__global__ void k_rowdot(const float* __restrict__ Xm, const float* __restrict__ a1,
                         const float* __restrict__ a2, float* __restrict__ s,
                         float* __restrict__ d, int M, int H)
{
    int lane = threadIdx.x & 31;
    int row  = blockIdx.x * (blockDim.x >> 5) + (threadIdx.x >> 5);
    if (row >= M) return;
    const float* xr = Xm + (size_t)row * H;
    float s1 = 0.f, s2 = 0.f;
    for (int f = lane; f < H; f += 32) {
        float xv = xr[f];
        s1 += xv * a1[f];
        if (a2) s2 += xv * a2[f];
    }
    s1 = wave_sum(s1);
    if (a2) s2 = wave_sum(s2);
    if (lane == 0) { s[row] = s1; if (a2) d[row] = s2; }
}

// alpha[e] = lrelu(s[src(e)] + d[dst(e)] (+ aE term), slope)
// src(e): srcA? (e<Elim? srcA[e] : e-Elim) : e   (null src => identity, for mol readout)
__global__ void k_alpha(const float* __restrict__ s, const float* __restrict__ d,
                        const int* __restrict__ srcA, const int* __restrict__ dstA,
                        const float* __restrict__ aE, const float* __restrict__ sumb,
                        float invE, int Elim, int Etot, float slope,
                        float* __restrict__ alpha)
{
    int e = blockIdx.x * blockDim.x + threadIdx.x;
    if (e >= Etot) return;
    int si = srcA ? ((e < Elim) ? srcA[e] : e - Elim) : e;
    int di = (e < Elim) ? dstA[e] : e - Elim;
    float a = s[si] + d[di];
    if (aE) a += (e < Elim) ? aE[e] : sumb[0] * invE;  // self loops use mean edge term
    alpha[e] = (a >= 0.f) ? a : slope * a;
}

__global__ void k_initmax(unsigned* __restrict__ mkey, int nseg)
{
    int i = blockIdx.x * blockDim.x + threadIdx.x;
    if (i < nseg) mkey[i] = 0x007FFFFFu;   // fkey(-inf)
}

__global__ void k_segmax(const float* __restrict__ alpha, const int* __restrict__ dstA,
                         int Elim, int Etot, unsigned* __restrict__ mkey)
{
    int e = blockIdx.x * blockDim.x + threadIdx.x;
    if (e >= Etot) return;
    int di = (e < Elim) ? dstA[e] : e - Elim;
    atomicMax(&mkey[di], fkey(alpha[e]));
}

__global__ void k_segexp(const float* __restrict__ alpha, const int* __restrict__ dstA,
                         int Elim, int Etot, const unsigned* __restrict__ mkey,
                         float* __restrict__ ssum, float* __restrict__ ew)
{
    int e = blockIdx.x * blockDim.x + threadIdx.x;
    if (e >= Etot) return;
    int di = (e < Elim) ? dstA[e] : e - Elim;
    float v = __expf(alpha[e] - fdec(mkey[di]));
    ew[e] = v;
    atomicAdd(&ssum[di], v);
}

// h[dst(e)][f] += feats[src(e)][f] * ew[e]/(ssum[dst]+eps)   (one block per edge, H threads)
__global__ void k_scatter(const float* __restrict__ feats, const int* __restrict__ srcA,
                          const int* __restrict__ dstA, int Elim,
                          const float* __restrict__ ew, const float* __restrict__ ssum,
                          float* __restrict__ hout, int H)
{
    int e = blockIdx.x;
    int f = threadIdx.x;
    int si = srcA ? ((e < Elim) ? srcA[e] : e - Elim) : e;
    int di = (e < Elim) ? dstA[e] : e - Elim;
    float w = ew[e] / (ssum[di] + 1e-16f);
    atomicAdd(&hout[(size_t)di * H + f], feats[(size_t)si * H + f] * w);
}

__global__ void k_bcast(float* __restrict__ dst, const float* __restrict__ bias,
                        int count, int H)
{
    int i = blockIdx.x * blockDim.x + threadIdx.x;
    if (i < count) dst[i] = bias[i % H];
}

__global__ void k_elu(float* __restrict__ b, int count)
{
    int i = blockIdx.x * blockDim.x + threadIdx.x;
    if (i >= count) return;
    float v = b[i];
    b[i] = (v > 0.f) ? v : (__expf(v) - 1.f);
}

__global__ void k_relu(float* __restrict__ b, int count)
{
    int i = blockIdx.x * blockDim.x + threadIdx.x;
    if (i >= count) return;
    float v = b[i];
    b[i] = (v > 0.f) ? v : 0.f;
}

// fused GRU gate + ReLU:  out = relu((1-z)*n + z*hx)
__global__ void k_gru(const float* __restrict__ gi, const float* __restrict__ gh,
                      const float* __restrict__ hx, float* __restrict__ out,
                      int M, int H)
{
    int idx = blockIdx.x * blockDim.x + threadIdx.x;
    if (idx >= M * H) return;
    int i = idx / H, f = idx % H;
    const float* gir = gi + (size_t)i * 3 * H;
    const float* ghr = gh + (size_t)i * 3 * H;
    float ir = gir[f], iz = gir[H + f], inn = gir[2 * H + f];
    float hr = ghr[f], hz = ghr[H + f], hn  = ghr[2 * H + f];
    float r = 1.f / (1.f + __expf(-(ir + hr)));
    float z = 1.f / (1.f + __expf(-(iz + hz)));
    float n = tanhf(inn + r * hn);
    float v = (1.f - z) * n + z * hx[idx];
    out[idx] = (v > 0.f) ? v : 0.f;
}

// global_add_pool: out[batch[i]][f] += xf[i][f]
__global__ void k_pool(const float* __restrict__ xf, const int* __restrict__ batch,
                       float* __restrict__ outg)
{
    int idx = blockIdx.x * blockDim.x + threadIdx.x;
    if (idx >= NN * HH) return;
    int i = idx / HH, f = idx % HH;
    atomicAdd(&outg[(size_t)batch[i] * HH + f], xf[idx]);
}

// y[g] = out[g] . W2 + b2   (one wave per graph)
__global__ void k_final(const float* __restrict__ outg, const float* __restrict__ W2,
                        const float* __restrict__ b2, float* __restrict__ y)
{
    int lane = threadIdx.x & 31;
    int g = blockIdx.x * (blockDim.x >> 5) + (threadIdx.x >> 5);
    if (g >= GG) return;
    const float* r = outg + (size_t)g * HH;
    float s = 0.f;
    for (int f = lane; f < HH; f += 32) s += r[f] * W2[f];
    s = wave_sum(s);
    if (lane == 0) y[g] = s + b2[0];
}

// ---------------- driver ----------------
extern "C" void kernel_launch(void* const* d_in, const int* in_sizes, int n_in,
                              void* d_out, int out_size, void* d_ws, size_t ws_size,
                              hipStream_t stream)
{
    (void)in_sizes; (void)n_in; (void)out_size; (void)ws_size;

    const float* x         = (const float*)d_in[0];
    const int*   ei        = (const int*)d_in[1];
    const float* edge_attr = (const float*)d_in[2];
    const int*   batch     = (const int*)d_in[3];
    const float* lin1_W = (const float*)d_in[4];
    const float* lin1_b = (const float*)d_in[5];
    const float* c0_W   = (const float*)d_in[6];
    const float* c0_as  = (const float*)d_in[7];
    const float* c0_ad  = (const float*)d_in[8];
    const float* c0_We  = (const float*)d_in[9];
    const float* c0_ae  = (const float*)d_in[10];
    const float* c0_b   = (const float*)d_in[11];
    const float* cv_W   = (const float*)d_in[12];
    const float* cv_as  = (const float*)d_in[13];
    const float* cv_ad  = (const float*)d_in[14];
    const float* cv_b   = (const float*)d_in[15];
    const float* g_Wih  = (const float*)d_in[16];
    const float* g_Whh  = (const float*)d_in[17];
    const float* g_bih  = (const float*)d_in[18];
    const float* g_bhh  = (const float*)d_in[19];
    const float* m_Wsrc = (const float*)d_in[20];
    const float* m_Wdst = (const float*)d_in[21];
    const float* m_as   = (const float*)d_in[22];
    const float* m_ad   = (const float*)d_in[23];
    const float* m_b    = (const float*)d_in[24];
    const float* mg_Wih = (const float*)d_in[25];
    const float* mg_Whh = (const float*)d_in[26];
    const float* mg_bih = (const float*)d_in[27];
    const float* mg_bhh = (const float*)d_in[28];
    const float* lin2_W = (const float*)d_in[29];
    const float* lin2_b = (const float*)d_in[30];

    const int* srcA = ei;
    const int* dstA = ei + EE_;

    // -------- workspace layout (floats) --------
    float* base = (float*)d_ws;
    size_t o = 0;
    float*    xf    = base + o; o += (size_t)NN * HH;        // node features
    float*    hbuf  = base + o; o += (size_t)NN * HH;        // GAT output h
    float*    Abuf  = base + o; o += (size_t)NN * 3 * HH;    // hW / gi / hs (phase-aliased)
    float*    Bbuf  = base + o; o += (size_t)NN * 3 * HH;    // gh
    float*    alpha = base + o; o += (size_t)(EE_ + NN);
    float*    ew    = base + o; o += (size_t)(EE_ + NN);
    float*    aE    = base + o; o += (size_t)EE_;
    float*    sbuf  = base + o; o += NN;
    float*    dbuf  = base + o; o += NN;
    unsigned* mkey  = (unsigned*)(base + o); o += NN;
    float*    ssum  = base + o; o += NN;
    float*    ve    = base + o; o += EDD;
    float*    sumb  = base + o; o += 8;
    float*    outg  = base + o; o += (size_t)GG * HH;
    float*    hdg   = base + o; o += (size_t)GG * HH;
    float*    hmol  = base + o; o += (size_t)GG * HH;
    float*    gim   = base + o; o += (size_t)GG * 3 * HH;
    float*    ghm   = base + o; o += (size_t)GG * 3 * HH;

    const int EB = 256;
    const int E0 = EE_ + NN;                                  // edges + self loops
    const int MT_N = NN / 16;                                 // 3125 m-tiles (N rows)
    const int MT_G = GG / 16;                                 // 128 m-tiles (G rows)
    dim3 gN_192((MT_N + 3) / 4, 192 / 64);   // 16x64 strip per wave
    dim3 gN_576((MT_N + 3) / 4, 576 / 64);
    dim3 gG_192((MT_G + 3) / 4, 192 / 64);
    dim3 gG_576((MT_G + 3) / 4, 576 / 64);

    // ---- lin1: xf = lrelu_0.01(x @ lin1_W^T + b) ----
    k_gemm_wmma<<<gN_192, 128, 0, stream>>>(x, lin1_W, lin1_b, xf, NN, CC, HH, 1);

    // ---- conv0 (GAT with edge features + self loops, slope 0.2) ----
    k_gemm_wmma<<<gN_192, 128, 0, stream>>>(xf, c0_W, nullptr, Abuf, NN, HH, HH, 0);
    k_wevec<<<1, 32, 0, stream>>>(c0_We, c0_ae, ve);
    hipMemsetAsync(sumb, 0, sizeof(float), stream);
    k_edgedot<<<(EE_ + EB - 1) / EB, EB, 0, stream>>>(edge_attr, ve, aE, sumb);
    k_rowdot<<<(NN + 7) / 8, 256, 0, stream>>>(Abuf, c0_as, c0_ad, sbuf, dbuf, NN, HH);
    k_alpha<<<(E0 + EB - 1) / EB, EB, 0, stream>>>(sbuf, dbuf, srcA, dstA, aE, sumb,
                                                   1.0f / EE_, EE_, E0, 0.2f, alpha);
    k_initmax<<<(NN + EB - 1) / EB, EB, 0, stream>>>(mkey, NN);
    k_segmax<<<(E0 + EB - 1) / EB, EB, 0, stream>>>(alpha, dstA, EE_, E0, mkey);
    hipMemsetAsync(ssum, 0, NN * sizeof(float), stream);
    k_segexp<<<(E0 + EB - 1) / EB, EB, 0, stream>>>(alpha, dstA, EE_, E0, mkey, ssum, ew);
    k_bcast<<<((size_t)NN * HH + EB - 1) / EB, EB, 0, stream>>>(hbuf, c0_b, NN * HH, HH);
    k_scatter<<<E0, HH, 0, stream>>>(Abuf, srcA, dstA, EE_, ew, ssum, hbuf, HH);
    k_elu<<<((size_t)NN * HH + EB - 1) / EB, EB, 0, stream>>>(hbuf, NN * HH);
    // GRU layer 0 (xf updated in place)
    k_gemm_wmma<<<gN_576, 128, 0, stream>>>(hbuf, g_Wih, g_bih, Abuf, NN, HH, 3 * HH, 0);
    k_gemm_wmma<<<gN_576, 128, 0, stream>>>(xf,   g_Whh, g_bhh, Bbuf, NN, HH, 3 * HH, 0);
    k_gru<<<((size_t)NN * HH + EB - 1) / EB, EB, 0, stream>>>(Abuf, Bbuf, xf, xf, NN, HH);

    // ---- remaining atom GAT + GRU layers (slope 0.01, no self loops) ----
    for (int l = 0; l < 2; ++l) {
        k_gemm_wmma<<<gN_192, 128, 0, stream>>>(xf, cv_W + (size_t)l * HH * HH,
                                                nullptr, Abuf, NN, HH, HH, 0);
        k_rowdot<<<(NN + 7) / 8, 256, 0, stream>>>(Abuf, cv_as + l * HH, cv_ad + l * HH,
                                                   sbuf, dbuf, NN, HH);
        k_alpha<<<(EE_ + EB - 1) / EB, EB, 0, stream>>>(sbuf, dbuf, srcA, dstA, nullptr,
                                                        nullptr, 0.f, EE_, EE_, 0.01f, alpha);
        k_initmax<<<(NN + EB - 1) / EB, EB, 0, stream>>>(mkey, NN);
        k_segmax<<<(EE_ + EB - 1) / EB, EB, 0, stream>>>(alpha, dstA, EE_, EE_, mkey);
        hipMemsetAsync(ssum, 0, NN * sizeof(float), stream);
        k_segexp<<<(EE_ + EB - 1) / EB, EB, 0, stream>>>(alpha, dstA, EE_, EE_, mkey, ssum, ew);
        k_bcast<<<((size_t)NN * HH + EB - 1) / EB, EB, 0, stream>>>(hbuf, cv_b + l * HH,
                                                                    NN * HH, HH);
        k_scatter<<<EE_, HH, 0, stream>>>(Abuf, srcA, dstA, EE_, ew, ssum, hbuf, HH);
        k_elu<<<((size_t)NN * HH + EB - 1) / EB, EB, 0, stream>>>(hbuf, NN * HH);
        const float* Wih = g_Wih + (size_t)(l + 1) * 3 * HH * HH;
        const float* Whh = g_Whh + (size_t)(l + 1) * 3 * HH * HH;
        k_gemm_wmma<<<gN_576, 128, 0, stream>>>(hbuf, Wih, g_bih + (l + 1) * 3 * HH,
                                                Abuf, NN, HH, 3 * HH, 0);
        k_gemm_wmma<<<gN_576, 128, 0, stream>>>(xf,   Whh, g_bhh + (l + 1) * 3 * HH,
                                                Bbuf, NN, HH, 3 * HH, 0);
        k_gru<<<((size_t)NN * HH + EB - 1) / EB, EB, 0, stream>>>(Abuf, Bbuf, xf, xf, NN, HH);
    }

    // ---- molecule readout ----
    hipMemsetAsync(outg, 0, (size_t)GG * HH * sizeof(float), stream);
    k_pool<<<((size_t)NN * HH + EB - 1) / EB, EB, 0, stream>>>(xf, batch, outg);
    k_relu<<<((size_t)GG * HH + EB - 1) / EB, EB, 0, stream>>>(outg, GG * HH);
    // hs = xf @ Wsrc^T   (fixed across timesteps; alias into Abuf)
    k_gemm_wmma<<<gN_192, 128, 0, stream>>>(xf, m_Wsrc, nullptr, Abuf, NN, HH, HH, 0);
    k_rowdot<<<(NN + 7) / 8, 256, 0, stream>>>(Abuf, m_as, nullptr, sbuf, nullptr, NN, HH);

    for (int t = 0; t < 2; ++t) {
        k_gemm_wmma<<<gG_192, 128, 0, stream>>>(outg, m_Wdst, nullptr, hdg, GG, HH, HH, 0);
        k_rowdot<<<(GG + 7) / 8, 256, 0, stream>>>(hdg, m_ad, nullptr, dbuf, nullptr, GG, HH);
        // node i -> molecule batch[i]; src is identity
        k_alpha<<<(NN + EB - 1) / EB, EB, 0, stream>>>(sbuf, dbuf, nullptr, batch, nullptr,
                                                       nullptr, 0.f, NN, NN, 0.01f, alpha);
        k_initmax<<<(GG + EB - 1) / EB, EB, 0, stream>>>(mkey, GG);
        k_segmax<<<(NN + EB - 1) / EB, EB, 0, stream>>>(alpha, batch, NN, NN, mkey);
        hipMemsetAsync(ssum, 0, GG * sizeof(float), stream);
        k_segexp<<<(NN + EB - 1) / EB, EB, 0, stream>>>(alpha, batch, NN, NN, mkey, ssum, ew);
        k_bcast<<<((size_t)GG * HH + EB - 1) / EB, EB, 0, stream>>>(hmol, m_b, GG * HH, HH);
        k_scatter<<<NN, HH, 0, stream>>>(Abuf, nullptr, batch, NN, ew, ssum, hmol, HH);
        k_elu<<<((size_t)GG * HH + EB - 1) / EB, EB, 0, stream>>>(hmol, GG * HH);
        k_gemm_wmma<<<gG_576, 128, 0, stream>>>(hmol, mg_Wih, mg_bih, gim, GG, HH, 3 * HH, 0);
        k_gemm_wmma<<<gG_576, 128, 0, stream>>>(outg, mg_Whh, mg_bhh, ghm, GG, HH, 3 * HH, 0);
        k_gru<<<((size_t)GG * HH + EB - 1) / EB, EB, 0, stream>>>(gim, ghm, outg, outg, GG, HH);
    }

    // ---- final linear ----
    k_final<<<(GG + 7) / 8, 256, 0, stream>>>(outg, lin2_W, lin2_b, (float*)d_out);
}